// FeaturesStructuringLayer_2207613190522
// MI455X (gfx1250) — compile-verified
//
#include <hip/hip_runtime.h>
#include <hip/hip_bf16.h>

// M = 262144 points, NF = 16 features, K = 85 kernel points (padded to 96).
// out[i] = [ F[i] (16) | FxQFT[i] (85) | (FxQFT @ QF)[i] (16) ]  -> 117 cols
//
// Fused plan:
//   phase1: QF_raw[96][16] = sum_i w[k,i] * F[i][:]   (fp32 WMMA 16x16x4, inner dim = points)
//   phase2: QF = omegaF * QF_raw ;  G = QF^T QF (16x16)  [since FxQFT@QF == F@(QF^T QF)]
//   phase3: per 16-pt tile: FxQFT = F_tile @ QF^T (6 wmma-N-tiles x 4 k-steps),
//           OUT2 = F_tile @ G (4 k-steps), plus copy of F.
//   Output stores are non-temporal (TH_NT): 122.8MB streamed once, never re-read,
//   keeps F (16MB) + QF/G resident in the 192MB L2 for the concurrent reads.

typedef float v2f __attribute__((ext_vector_type(2)));
typedef float v8f __attribute__((ext_vector_type(8)));

#define KPAD 96
#define NF   16

__device__ __forceinline__ v8f wmma4(v2f a, v2f b, v8f c) {
    // V_WMMA_F32_16X16X4_F32: D = A(16x4) x B(4x16) + C(16x16), fp32
    return __builtin_amdgcn_wmma_f32_16x16x4_f32(
        /*neg_a=*/false, a, /*neg_b=*/false, b,
        /*c_mod=*/(short)0, c, /*reuse_a=*/false, /*reuse_b=*/false);
}

__device__ __forceinline__ v8f v8zero() {
    v8f z;
#pragma unroll
    for (int j = 0; j < 8; ++j) z[j] = 0.0f;
    return z;
}

// ---------------------------------------------------------------- zero ws ---
__global__ void k_zero_ws(float* __restrict__ ws) {
    int i = blockIdx.x * blockDim.x + threadIdx.x;
    if (i < KPAD * NF) ws[i] = 0.0f;
}

// ------------------------------------------------------------------ phase1 ---
// QF_raw[k][f] += exp(||X[p]-QX[k]||^2 / wD[k]^2) * F[p][f], reduced over all M.
// One wave processes chunks of 4 points; A = weights (16 k-rows x 4 points),
// B = F rows (4 points x 16 features); 6 k-tiles cover K=96 (rows >=85 masked 0).
__global__ void __launch_bounds__(256) k_phase1(
    const float* __restrict__ X, const float* __restrict__ F,
    const float* __restrict__ QX, const float* __restrict__ omegaD,
    float* __restrict__ wsRaw, int M)
{
    __shared__ float sQX[KPAD * 3];
    __shared__ float sIW[KPAD];
    __shared__ float sAcc[KPAD * NF];

    const int tid = threadIdx.x;
    if (tid < KPAD) {
        if (tid < 85) {
            sQX[tid * 3 + 0] = QX[tid * 3 + 0];
            sQX[tid * 3 + 1] = QX[tid * 3 + 1];
            sQX[tid * 3 + 2] = QX[tid * 3 + 2];
            float od = omegaD[tid];
            sIW[tid] = 1.0f / (od * od);
        } else {
            sQX[tid * 3 + 0] = 0.0f; sQX[tid * 3 + 1] = 0.0f; sQX[tid * 3 + 2] = 0.0f;
            sIW[tid] = 0.0f;
        }
    }
    for (int i = tid; i < KPAD * NF; i += blockDim.x) sAcc[i] = 0.0f;
    __syncthreads();

    const int lane = tid & 31;
    const int half = lane >> 4;   // 0: points {0,1} of chunk, 1: points {2,3}
    const int col  = lane & 15;   // k-row within tile / feature column
    const int wib  = tid >> 5;
    const int wpb  = blockDim.x >> 5;
    const int gwave  = blockIdx.x * wpb + wib;
    const int nwaves = gridDim.x * wpb;
    const int nchunks = M >> 2;

    v8f acc[6];
#pragma unroll
    for (int t = 0; t < 6; ++t) acc[t] = v8zero();

    for (int c = gwave; c < nchunks; c += nwaves) {
        const int p = (c << 2) + (half << 1);  // this lane's first point
        const float x0 = X[p * 3 + 0], y0 = X[p * 3 + 1], z0 = X[p * 3 + 2];
        const float x1 = X[p * 3 + 3], y1 = X[p * 3 + 4], z1 = X[p * 3 + 5];
        v2f b;                                  // B: F[point][feature=col]
        b.x = F[(size_t)p * NF + col];
        b.y = F[(size_t)(p + 1) * NF + col];
#pragma unroll
        for (int t = 0; t < 6; ++t) {
            const int k = t * 16 + col;
            const float qx = sQX[k * 3 + 0], qy = sQX[k * 3 + 1], qz = sQX[k * 3 + 2];
            const float iw = sIW[k];
            const float dx0 = x0 - qx, dy0 = y0 - qy, dz0 = z0 - qz;
            const float dx1 = x1 - qx, dy1 = y1 - qy, dz1 = z1 - qz;
            const float d0 = dx0 * dx0 + dy0 * dy0 + dz0 * dz0;
            const float d1 = dx1 * dx1 + dy1 * dy1 + dz1 * dz1;
            const float m = (k < 85) ? 1.0f : 0.0f;
            v2f a;
            a.x = m * __expf(d0 * iw);
            a.y = m * __expf(d1 * iw);
            acc[t] = wmma4(a, b, acc[t]);
        }
    }

    // wave partials -> block LDS accumulator
#pragma unroll
    for (int t = 0; t < 6; ++t) {
#pragma unroll
        for (int v = 0; v < 8; ++v) {
            const int k = t * 16 + v + half * 8;   // C layout: row = v + 8*(lane/16)
            unsafeAtomicAdd(&sAcc[k * NF + col], acc[t][v]);
        }
    }
    __syncthreads();
    for (int i = tid; i < KPAD * NF; i += blockDim.x)
        unsafeAtomicAdd(&wsRaw[i], sAcc[i]);
}

// ------------------------------------------------------------------ phase2 ---
// QF = omegaF * QF_raw ; G = QF^T QF (16x16). One block, 256 threads.
__global__ void __launch_bounds__(256) k_phase2(
    const float* __restrict__ wsRaw, const float* __restrict__ omegaF,
    float* __restrict__ wsQF, float* __restrict__ wsG)
{
    __shared__ float sQF[KPAD * NF];
    const int tid = threadIdx.x;
    for (int i = tid; i < KPAD * NF; i += 256) {
        const float q = wsRaw[i] * omegaF[i & 15];
        sQF[i] = q;
        wsQF[i] = q;
    }
    __syncthreads();
    const int f1 = tid >> 4, f2 = tid & 15;
    float s = 0.0f;
#pragma unroll 8
    for (int k = 0; k < KPAD; ++k) s += sQF[k * NF + f1] * sQF[k * NF + f2];
    wsG[f1 * NF + f2] = s;
}

// ------------------------------------------------------------------ phase3 ---
// Per 16-point tile: FxQFT (6 wmma tiles x 4 k-steps) + F@G (4 k-steps) + copy F.
// All output stores are non-temporal (streaming write, no reuse).
__global__ void __launch_bounds__(256) k_phase3(
    const float* __restrict__ F, const float* __restrict__ wsQF,
    const float* __restrict__ wsG, float* __restrict__ out, int M)
{
    const int tid  = threadIdx.x;
    const int lane = tid & 31;
    const int half = lane >> 4;
    const int col  = lane & 15;
    const int wib  = tid >> 5;
    const int wpb  = blockDim.x >> 5;
    const int gwave  = blockIdx.x * wpb + wib;
    const int nwaves = gridDim.x * wpb;
    const int ntiles = M >> 4;

    // Hoist B fragments (loop-invariant): QF^T for stage1, G for stage2.
    // B layout mirror of A: vgpr0 -> inner = c*4 + 2*half, vgpr1 -> +1; n = col.
    v2f Bq[6][4];
#pragma unroll
    for (int t = 0; t < 6; ++t) {
#pragma unroll
        for (int c = 0; c < 4; ++c) {
            const int f = c * 4 + half * 2;               // inner = feature
            Bq[t][c].x = wsQF[(t * 16 + col) * NF + f];   // B[f][k=col] = QF[k][f]
            Bq[t][c].y = wsQF[(t * 16 + col) * NF + f + 1];
        }
    }
    v2f Bg[4];
#pragma unroll
    for (int c = 0; c < 4; ++c) {
        const int kin = c * 4 + half * 2;                 // inner = feature (G index)
        Bg[c].x = wsG[kin * NF + col];
        Bg[c].y = wsG[(kin + 1) * NF + col];
    }

    for (int tile = gwave; tile < ntiles; tile += nwaves) {
        const int base = tile << 4;

        // A fragments: A[m=col][inner feature]; lane reads its point's row.
        const float* frow = F + (size_t)(base + col) * NF;
        v2f Af[4];
#pragma unroll
        for (int c = 0; c < 4; ++c) {
            const int f = c * 4 + half * 2;
            Af[c].x = frow[f];
            Af[c].y = frow[f + 1];
        }

        // ---- stage 1: FxQFT = F_tile @ QF^T, store cols 16..100 (k<85)
#pragma unroll
        for (int t = 0; t < 6; ++t) {
            v8f acc = v8zero();
#pragma unroll
            for (int c = 0; c < 4; ++c) acc = wmma4(Af[c], Bq[t][c], acc);
            const int k = t * 16 + col;
            if (k < 85) {
#pragma unroll
                for (int v = 0; v < 8; ++v) {
                    const int m = v + half * 8;
                    __builtin_nontemporal_store(
                        acc[v], &out[(size_t)(base + m) * 117 + 16 + k]);
                }
            }
        }

        // ---- stage 2: OUT2 = F_tile @ G, store cols 101..116
        v8f acc2 = v8zero();
#pragma unroll
        for (int c = 0; c < 4; ++c) acc2 = wmma4(Af[c], Bg[c], acc2);
#pragma unroll
        for (int v = 0; v < 8; ++v) {
            const int m = v + half * 8;
            __builtin_nontemporal_store(
                acc2[v], &out[(size_t)(base + m) * 117 + 101 + col]);
        }

        // ---- copy F into cols 0..15
#pragma unroll
        for (int rr = 0; rr < 8; ++rr) {
            const int p = rr * 2 + half;
            __builtin_nontemporal_store(
                F[(size_t)(base + p) * NF + col],
                &out[(size_t)(base + p) * 117 + col]);
        }
    }
}

// ------------------------------------------------------------------ launch ---
extern "C" void kernel_launch(void* const* d_in, const int* in_sizes, int n_in,
                              void* d_out, int out_size, void* d_ws, size_t ws_size,
                              hipStream_t stream) {
    const float* X      = (const float*)d_in[0];
    const float* F      = (const float*)d_in[1];
    const float* QX     = (const float*)d_in[2];
    const float* omegaD = (const float*)d_in[3];
    const float* omegaF = (const float*)d_in[4];
    float* out = (float*)d_out;
    float* ws  = (float*)d_ws;

    const int M = in_sizes[0] / 3;

    float* wsRaw = ws;                 // [96*16]
    float* wsQF  = ws + KPAD * NF;     // [96*16]
    float* wsG   = ws + 2 * KPAD * NF; // [16*16]

    k_zero_ws<<<(KPAD * NF + 255) / 256, 256, 0, stream>>>(wsRaw);
    k_phase1<<<256, 256, 0, stream>>>(X, F, QX, omegaD, wsRaw, M);
    k_phase2<<<1, 256, 0, stream>>>(wsRaw, omegaF, wsQF, wsG);
    k_phase3<<<512, 256, 0, stream>>>(F, wsQF, wsG, out, M);
}